// AngularHallucination_33964601376960
// MI455X (gfx1250) — compile-verified
//
#include <hip/hip_runtime.h>
#include <hip/hip_bf16.h>
#include <math.h>

typedef __attribute__((ext_vector_type(2))) float v2f;
typedef __attribute__((ext_vector_type(8))) float v8f;

#define B_DIM 8
#define C_DIM 64
#define H_DIM 256
#define W_DIM 256
#define N_DIM (H_DIM * W_DIM)
#define CHUNKS 32
#define ROWS_PER_CHUNK (H_DIM / CHUNKS)
#define XSTR 260   // padded LDS row stride (floats), multiple of 4, bank stride 4
#define PSTR 68    // padded LDS row stride for kernel 3

// ---------------------------------------------------------------------------
// Kernel 1: fused grid-sample (reduced to 1-D x-interpolation) + partial
//           corr[b,i,j] = sum_n x[b,i,n] * xa[b,j,n]  via V_WMMA_F32_16X16X4_F32
// grid = (B, CHUNKS), block = 512 (16 waves); each wave owns one 16x16 tile.
// ---------------------------------------------------------------------------
__global__ __launch_bounds__(512, 1)
void corr_partial_kernel(const float* __restrict__ x,
                         const int* __restrict__ angle_p,
                         float* __restrict__ partials /* [B][CHUNKS][64][64] */)
{
    extern __shared__ float smem[];
    float* xl   = smem;                    // [64][XSTR]
    float* xal  = smem + 64 * XSTR;        // [64][XSTR]
    float* fx_s = smem + 2 * 64 * XSTR;    // [256]
    float* wz_s = fx_s + 256;              // [256]
    int*   ix0_s = (int*)(wz_s + 256);     // [256]

    const int b     = blockIdx.x;
    const int chunk = blockIdx.y;
    const int tid   = threadIdx.x;
    const int lane  = tid & 31;
    const int wave  = tid >> 5;
    const int it    = wave >> 2;           // i-tile 0..3
    const int jt    = wave & 3;            // j-tile 0..3
    const int mrow  = lane & 15;
    const int ksel  = (lane >> 4) << 1;    // 0 or 2

    // Per-w interpolation params (angle in *radians* per the reference quirk)
    if (tid < W_DIM) {
        const float ang = (float)angle_p[0];
        const float cth = cosf(ang);
        const float sth = sinf(ang);
        const int w = tid;
        const float gx = (2.0f * (float)w + 1.0f) / (float)W_DIM - 1.0f;
        // ix = ((c*gx + 1)*W - 1)/2   (align_corners=False)
        const float ix = ((cth * gx + 1.0f) * (float)W_DIM - 1.0f) * 0.5f;
        // iz = ((-s*gx + 1)*D - 1)/2 with D=1
        const float iz = (-sth * gx) * 0.5f;
        const float ix0f = floorf(ix);
        const float iz0f = floorf(iz);
        const float fx = ix - ix0f;
        const float fz = iz - iz0f;
        const int ix0 = (int)ix0f;
        const int iz0 = (int)iz0f;
        // only zi==0 can be valid (D==1)
        float wz = 0.0f;
        if (iz0 == 0)       wz = 1.0f - fz;
        else if (iz0 == -1) wz = fz;
        ix0_s[w] = ix0;
        fx_s[w]  = fx;
        wz_s[w]  = wz;
    }

    v8f acc = {};

    for (int r = 0; r < ROWS_PER_CHUNK; ++r) {
        const int h = chunk * ROWS_PER_CHUNK + r;
        __syncthreads();   // protect LDS from previous iteration's readers

        // stage x[b, :, h, :]  (64 x 256 floats) as float4
        for (int v = tid; v < C_DIM * (W_DIM / 4); v += 512) {
            const int cch = v >> 6;           // channel
            const int w4  = v & 63;           // float4 index in row
            const float4* src = (const float4*)(x +
                (((size_t)b * C_DIM + cch) * H_DIM + h) * W_DIM);
            *((float4*)&xl[cch * XSTR + w4 * 4]) = src[w4];
        }
        __syncthreads();

        // compute xa row: xa[c][w] = wz*( (1-fx)*x[c][ix0] + fx*x[c][ix0+1] )
        for (int v = tid; v < C_DIM * W_DIM; v += 512) {
            const int cch = v >> 8;
            const int w   = v & 255;
            const int i0  = ix0_s[w];
            const float f = fx_s[w];
            const float wz = wz_s[w];
            const float v0 = (i0 >= 0 && i0 < W_DIM)         ? xl[cch * XSTR + i0]     : 0.0f;
            const float v1 = (i0 + 1 >= 0 && i0 + 1 < W_DIM) ? xl[cch * XSTR + i0 + 1] : 0.0f;
            xal[cch * XSTR + w] = wz * ((1.0f - f) * v0 + f * v1);
        }
        __syncthreads();

        // WMMA: D(16x16) += A(16x4) x B(4x16), K sweep over 256
        const int arow = it * 16 + mrow;   // A: M = lane%16
        const int brow = jt * 16 + mrow;   // B[K][N]: N = lane%16 -> xa channel
        for (int k = 0; k < W_DIM; k += 4) {
            v2f a  = *(const v2f*)&xl[arow * XSTR + k + ksel];
            v2f bb = *(const v2f*)&xal[brow * XSTR + k + ksel];
            acc = __builtin_amdgcn_wmma_f32_16x16x4_f32(
                false, a, false, bb, (short)0, acc, false, false);
        }
    }

    // store 16x16 partial tile (D layout: VGPR r -> M=r+8*(lane/16), N=lane%16)
    float* dst = partials + (((size_t)b * CHUNKS + chunk) * C_DIM * C_DIM);
    const int ncol  = lane & 15;
    const int mbase = (lane >> 4) * 8;
    #pragma unroll
    for (int r = 0; r < 8; ++r) {
        const int mi = it * 16 + mbase + r;
        const int nj = jt * 16 + ncol;
        dst[mi * C_DIM + nj] = acc[r];
    }
}

// ---------------------------------------------------------------------------
// Kernel 2: deterministic reduction of partials + softmax over dim=1 (i)
// grid = B, block = 256
// ---------------------------------------------------------------------------
__global__ __launch_bounds__(256, 1)
void reduce_softmax_kernel(const float* __restrict__ partials,
                           float* __restrict__ pro /* [B][64][64] */)
{
    __shared__ float corr[C_DIM * C_DIM];
    const int b = blockIdx.x;
    const int tid = threadIdx.x;

    for (int e = tid; e < C_DIM * C_DIM; e += 256) {
        float s = 0.0f;
        for (int c = 0; c < CHUNKS; ++c)
            s += partials[(((size_t)b * CHUNKS + c) * C_DIM * C_DIM) + e];
        corr[e] = s;
    }
    __syncthreads();

    if (tid < C_DIM) {
        const int j = tid;
        float m = -INFINITY;
        for (int i = 0; i < C_DIM; ++i)
            m = fmaxf(m, corr[i * C_DIM + j]);
        float sum = 0.0f;
        for (int i = 0; i < C_DIM; ++i)
            sum += __expf(corr[i * C_DIM + j] - m);
        const float inv = 1.0f / sum;
        for (int i = 0; i < C_DIM; ++i)
            pro[((size_t)b * C_DIM + i) * C_DIM + j] =
                __expf(corr[i * C_DIM + j] - m) * inv;
    }
}

// ---------------------------------------------------------------------------
// Kernel 3: out[b,i,n] = sum_j pro[b,i,j] * x[b,j,n]   (WMMA, K=64)
// grid = (N/64, B), block = 512 (16 waves); each wave one 16x16 (i x n) tile.
// ---------------------------------------------------------------------------
__global__ __launch_bounds__(512, 1)
void apply_pro_kernel(const float* __restrict__ pro,
                      const float* __restrict__ x,
                      float* __restrict__ out)
{
    __shared__ float pro_lds[C_DIM * PSTR];  // [i][j]
    __shared__ float x_lds[C_DIM * PSTR];    // [j][n]

    const int b  = blockIdx.y;
    const int n0 = blockIdx.x * 64;
    const int tid  = threadIdx.x;
    const int lane = tid & 31;
    const int wave = tid >> 5;
    const int it = wave >> 2;                // i-tile 0..3
    const int nt = wave & 3;                 // n-tile 0..3
    const int mrow = lane & 15;
    const int ksel = (lane >> 4) << 1;

    // stage pro[b] (64x64)
    for (int v = tid; v < C_DIM * C_DIM; v += 512) {
        const int i = v >> 6, j = v & 63;
        pro_lds[i * PSTR + j] = pro[(size_t)b * C_DIM * C_DIM + v];
    }
    // stage x[b, :, n0:n0+64] as float4
    for (int v = tid; v < C_DIM * 16; v += 512) {
        const int cch = v >> 4;
        const int q   = v & 15;
        const float4* src = (const float4*)(x +
            ((size_t)b * C_DIM + cch) * (size_t)N_DIM + n0);
        *((float4*)&x_lds[cch * PSTR + q * 4]) = src[q];
    }
    __syncthreads();

    v8f acc = {};
    const int arow = it * 16 + mrow;         // A: M = lane%16 (i)
    const int ncol = nt * 16 + mrow;         // B: N = lane%16 (n)
    for (int jj = 0; jj < C_DIM; jj += 4) {
        v2f a = *(const v2f*)&pro_lds[arow * PSTR + jj + ksel];
        v2f bb;
        bb.x = x_lds[(jj + ksel + 0) * PSTR + ncol];
        bb.y = x_lds[(jj + ksel + 1) * PSTR + ncol];
        acc = __builtin_amdgcn_wmma_f32_16x16x4_f32(
            false, a, false, bb, (short)0, acc, false, false);
    }

    const int nc    = lane & 15;
    const int mbase = (lane >> 4) * 8;
    #pragma unroll
    for (int r = 0; r < 8; ++r) {
        const int mi = it * 16 + mbase + r;
        const int n  = n0 + nt * 16 + nc;
        out[((size_t)b * C_DIM + mi) * (size_t)N_DIM + n] = acc[r];
    }
}

// ---------------------------------------------------------------------------
extern "C" void kernel_launch(void* const* d_in, const int* in_sizes, int n_in,
                              void* d_out, int out_size, void* d_ws, size_t ws_size,
                              hipStream_t stream)
{
    const float* x     = (const float*)d_in[0];
    const int*   angle = (const int*)d_in[1];
    float* out = (float*)d_out;

    float* partials = (float*)d_ws;                              // 8*32*64*64 floats = 4 MB
    float* pro      = partials + (size_t)B_DIM * CHUNKS * C_DIM * C_DIM; // 8*64*64 floats

    const size_t smem1 = (2 * 64 * XSTR + 3 * 256) * sizeof(float);

    dim3 g1(B_DIM, CHUNKS);
    corr_partial_kernel<<<g1, 512, smem1, stream>>>(x, angle, partials);

    reduce_softmax_kernel<<<B_DIM, 256, 0, stream>>>(partials, pro);

    dim3 g3(N_DIM / 64, B_DIM);
    apply_pro_kernel<<<g3, 512, 0, stream>>>(pro, x, out);
}